// KMeansClusteringBaseline_46600395162216
// MI455X (gfx1250) — compile-verified
//
#include <hip/hip_runtime.h>

// ---------------------------------------------------------------------------
// Batched k-means for x:(32, 20000, 2) f32, K=256, 10 iterations.
// Assignment computes the FULL squared distance inside one WMMA:
//   A[m] = (px, py, p2, 1)   B[n] = (-2cx, -2cy, 1, c2)   C = 0
//   D[m][n] = p2 - 2*(px*cx+py*cy) + c2  >= 0 (up to rounding)
// Argmin via packed sortable keys compared as SIGNED ints:
//   key = (bits(dist) & ~0xFF) | centroid_index,  best = v_min_i32
// Non-negative floats order correctly as signed ints; a -eps rounding of a
// zero distance gives a very negative key that (correctly) wins the min.
// Per candidate: v_and_or_b32 + v_min_i32 only. 16 B-tiles in registers,
// 16-WMMA sweep fully unrolled.
// ---------------------------------------------------------------------------

#define KC    256   // clusters
#define BATCH 32    // batch size
#define ITERS 10

typedef __attribute__((ext_vector_type(2))) float v2f;
typedef __attribute__((ext_vector_type(8))) float v8f;

// Deterministic evenly-strided centroid init (stand-in for randperm init).
__global__ void kmeans_init_centroids(const float* __restrict__ x,
                                      float* __restrict__ cent, int n) {
  int t = blockIdx.x * blockDim.x + threadIdx.x;
  if (t >= BATCH * KC) return;
  int b = t / KC, k = t % KC;
  long long src = ((long long)b * n + (long long)((long long)k * n / KC)) * 2;
  cent[t * 2 + 0] = x[src + 0];
  cent[t * 2 + 1] = x[src + 1];
}

__global__ void kmeans_zero(float* __restrict__ sums, float* __restrict__ cnts) {
  int t = blockIdx.x * blockDim.x + threadIdx.x;
  if (t < BATCH * KC) {
    sums[t * 2 + 0] = 0.f;
    sums[t * 2 + 1] = 0.f;
    cnts[t]         = 0.f;
  }
}

// Fused assignment + cluster accumulation. One block = 8 waves, grid-strided
// over 16-point tiles of one batch row. Partial sums accumulate in LDS
// (ds_add_f32) and flush with one global atomic per cluster per block.
__global__ void __launch_bounds__(256)
kmeans_assign(const float* __restrict__ x, const float* __restrict__ cent,
              int* __restrict__ ids, float* __restrict__ sums,
              float* __restrict__ cnts, int n) {
  __shared__ float s_cx[KC], s_cy[KC], s_c2[KC];   // cx,cy pre-scaled by -2
  __shared__ float s_sx[KC], s_sy[KC], s_cn[KC];

  const int b   = blockIdx.y;
  const int tid = threadIdx.x;

  {
    float cx = cent[(b * KC + tid) * 2 + 0];
    float cy = cent[(b * KC + tid) * 2 + 1];
    s_cx[tid] = -2.0f * cx;
    s_cy[tid] = -2.0f * cy;
    s_c2[tid] = cx * cx + cy * cy;
    s_sx[tid] = 0.f;
    s_sy[tid] = 0.f;
    s_cn[tid] = 0.f;
  }
  __syncthreads();

  const int lane = tid & 31;
  const int wave = tid >> 5;    // 0..7
  const int half = lane >> 4;   // 0: lanes carry K=0,1 ; 1: lanes carry K=2,3
  const int nl   = lane & 15;
  const float* __restrict__ xb = x + (size_t)b * n * 2;

  // Hoist all 16 B-tiles (4x16 f32) into registers, loop-invariant:
  //   rows K=0,1 = (-2cx, -2cy), rows K=2,3 = (1, c2)
  v2f Bm[16];
#pragma unroll
  for (int ct = 0; ct < 16; ++ct) {
    float cx = s_cx[ct * 16 + nl];
    float cy = s_cy[ct * 16 + nl];
    float c2 = s_c2[ct * 16 + nl];
    Bm[ct].x = half ? 1.0f : cx;
    Bm[ct].y = half ? c2   : cy;
  }

  const int ntiles = (n + 15) >> 4;
  for (int tile = blockIdx.x * 8 + wave; tile < ntiles; tile += gridDim.x * 8) {
    const int m0 = tile << 4;

    // Every lane loads point m0+nl; lanes 0-15 feed (px,py) into K=0,1,
    // lanes 16-31 feed (p2, 1) into K=2,3. Branch-free: EXEC all-1s at WMMAs.
    int   mload = m0 + nl;
    int   mc    = mload < n ? mload : n - 1;     // clamped load address
    float lx    = xb[mc * 2 + 0];
    float ly    = xb[mc * 2 + 1];
    bool  valid = (mload < n);
    float p2    = lx * lx + ly * ly;
    v2f A;
    A.x = valid ? (half ? p2   : lx) : 0.f;
    A.y = valid ? (half ? 1.0f : ly) : 0.f;

    int best[8];
#pragma unroll
    for (int r = 0; r < 8; ++r) best[r] = 0x7FFFFFFF;

    // 16 centroid tiles of 16 -> full K=256 sweep; D is the squared distance.
#pragma unroll
    for (int ct = 0; ct < 16; ++ct) {
      v8f C = {};
      v8f D = __builtin_amdgcn_wmma_f32_16x16x4_f32(
          /*neg_a=*/false, A, /*neg_b=*/false, Bm[ct],
          /*c_mod=*/(short)0, C, /*reuse_a=*/false, /*reuse_b=*/false);

#pragma unroll
      for (int r = 0; r < 8; ++r) {
        // v_and_or_b32 + v_min_i32: dist bits (>=0) order as signed ints.
        int kb = (int)((__float_as_uint(D[r]) & 0xFFFFFF00u) |
                       (unsigned)(ct << 4));
        best[r] = kb < best[r] ? kb : best[r];
      }
    }
    // nl is constant per lane -> OR once; keys now carry centroid idx = ct*16+nl.
#pragma unroll
    for (int r = 0; r < 8; ++r) best[r] |= nl;

    // Cross-lane argmin inside each 16-lane half (xor butterfly on i32 keys).
#pragma unroll
    for (int off = 1; off < 16; off <<= 1) {
#pragma unroll
      for (int r = 0; r < 8; ++r) {
        int ob = __shfl_xor(best[r], off, 32);
        best[r] = ob < best[r] ? ob : best[r];
      }
    }

    // Lane nl==r of each half writes point m = m0 + r + 8*half; its coords
    // were loaded by lane r + 8*half -- fetch via shuffle, no global re-load.
#pragma unroll
    for (int r = 0; r < 8; ++r) {
      float px = __shfl(lx, r + 8 * half, 32);
      float py = __shfl(ly, r + 8 * half, 32);
      if (nl == r) {
        int m = m0 + r + 8 * half;
        if (m < n) {
          int bi = best[r] & 0xFF;
          ids[(size_t)b * n + m] = bi;
          atomicAdd(&s_sx[bi], px);
          atomicAdd(&s_sy[bi], py);
          atomicAdd(&s_cn[bi], 1.0f);
        }
      }
    }
  }

  __syncthreads();
  atomicAdd(&sums[(b * KC + tid) * 2 + 0], s_sx[tid]);
  atomicAdd(&sums[(b * KC + tid) * 2 + 1], s_sy[tid]);
  atomicAdd(&cnts[b * KC + tid], s_cn[tid]);
}

// centroid <- sum/count, keep old centroid for empty clusters.
__global__ void kmeans_update(float* __restrict__ cent,
                              const float* __restrict__ sums,
                              const float* __restrict__ cnts) {
  int b = blockIdx.x, k = threadIdx.x;
  float c = cnts[b * KC + k];
  if (c > 0.f) {
    cent[(b * KC + k) * 2 + 0] = sums[(b * KC + k) * 2 + 0] / c;
    cent[(b * KC + k) * 2 + 1] = sums[(b * KC + k) * 2 + 1] / c;
  }
}

extern "C" void kernel_launch(void* const* d_in, const int* in_sizes, int n_in,
                              void* d_out, int out_size, void* d_ws, size_t ws_size,
                              hipStream_t stream) {
  const float* x = (const float*)d_in[0];
  const int n = in_sizes[0] / (BATCH * 2);   // 20000
  int* ids = (int*)d_out;

  // workspace: centroids | sums | counts (all f32)
  float* cent = (float*)d_ws;                 // BATCH*KC*2
  float* sums = cent + BATCH * KC * 2;        // BATCH*KC*2
  float* cnts = sums + BATCH * KC * 2;        // BATCH*KC

  kmeans_init_centroids<<<(BATCH * KC + 255) / 256, 256, 0, stream>>>(x, cent, n);

  dim3 agrid(40, BATCH);   // 40 blocks/batch * 8 waves = 320 waves per batch row
  for (int it = 0; it < ITERS; ++it) {
    kmeans_zero<<<(BATCH * KC + 255) / 256, 256, 0, stream>>>(sums, cnts);
    kmeans_assign<<<agrid, 256, 0, stream>>>(x, cent, ids, sums, cnts, n);
    kmeans_update<<<BATCH, KC, 0, stream>>>(cent, sums, cnts);
  }
}